// VQVAEMnist_35983236006000
// MI455X (gfx1250) — compile-verified
//
#include <hip/hip_runtime.h>
#include <math.h>

typedef __attribute__((ext_vector_type(2))) float v2f;
typedef __attribute__((ext_vector_type(8))) float v8f;

// D = A(16x4) * B(4x16) + C, fp32 throughout.
// 8-arg form: (neg_a, A, neg_b, B, c_mod, C, reuse_a, reuse_b)
__device__ __forceinline__ v8f wmma4(v2f a, v2f b, v8f c) {
    return __builtin_amdgcn_wmma_f32_16x16x4_f32(false, a, false, b, (short)0, c,
                                                 false, false);
}

// ---------------------------------------------------------------------------
// A-element fetch for implicit-GEMM conv. k = ic*9 + ky*3 + kx.
// If UPS: logical input is 2x bilinear upsample (half-pixel centers, edge
// clamp -- matches jax.image.resize 'bilinear') of the stored INHxINW tensor.
// ---------------------------------------------------------------------------
template <int CIN, int STRIDE, int INH, int INW, bool UPS>
__device__ __forceinline__ float fetchA(const float* __restrict__ in, int bb,
                                        int k, int oy, int ox) {
    constexpr int LH = UPS ? INH * 2 : INH;
    constexpr int LW = UPS ? INW * 2 : INW;
    const int ic = (unsigned)k / 9u;
    const int t  = k - ic * 9;
    const int ky = (unsigned)t / 3u;
    const int kx = t - ky * 3;
    const int iy = oy * STRIDE + ky - 1;
    const int ix = ox * STRIDE + kx - 1;
    if (iy < 0 || iy >= LH || ix < 0 || ix >= LW) return 0.0f;
    const float* p = in + ((size_t)bb * CIN + ic) * ((size_t)INH * INW);
    if (!UPS) return p[iy * INW + ix];
    const float sy = iy * 0.5f - 0.25f;
    const float sx = ix * 0.5f - 0.25f;
    const int y0 = (int)floorf(sy);
    const int x0 = (int)floorf(sx);
    const float fy = sy - (float)y0;
    const float fx = sx - (float)x0;
    const int y0c = min(max(y0, 0), INH - 1);
    const int y1c = min(max(y0 + 1, 0), INH - 1);
    const int x0c = min(max(x0, 0), INW - 1);
    const int x1c = min(max(x0 + 1, 0), INW - 1);
    const float v00 = p[y0c * INW + x0c];
    const float v01 = p[y0c * INW + x1c];
    const float v10 = p[y1c * INW + x0c];
    const float v11 = p[y1c * INW + x1c];
    const float a0 = v00 + (v01 - v00) * fx;
    const float a1 = v10 + (v11 - v10) * fx;
    return a0 + (a1 - a0) * fy;
}

// ---------------------------------------------------------------------------
// 3x3 conv (pad=1) as implicit GEMM via V_WMMA_F32_16X16X4_F32.
// One wave owns 16 output pixels (one row-tile) x all output channels.
// Weights OIHW: B[k, oc] = w[oc*CIN*9 + k]; staged into LDS in 32KB chunks as
// k-pairs so the hot loop reads one ds_load_b64 per WMMA B-operand.
// ---------------------------------------------------------------------------
template <int CIN, int COUT, int STRIDE, int INH, int INW, int OH, int OW,
          bool UPS, bool RELU, int CVALID>
__global__ __launch_bounds__(256) void conv3x3_wmma(
    const float* __restrict__ in, const float* __restrict__ w,
    const float* __restrict__ bias, float* __restrict__ out, int nTiles) {
    constexpr int KTOT  = CIN * 9;
    constexpr int KP    = (KTOT + 3) & ~3;  // pad K to multiple of 4
    constexpr int NOC   = COUT / 16;
    constexpr int PAIRS = KP / 2;                 // total k-pairs
    constexpr int CHUNK = 4096 / COUT;            // pairs per 32KB LDS chunk
    __shared__ v2f spair[4096];                   // [pair][oc], 32 KB

    const int lane = threadIdx.x & 31;
    const int waveR = blockIdx.x * (blockDim.x >> 5) + (threadIdx.x >> 5);
    const int wave  = min(waveR, nTiles - 1);     // clamp (all waves must sync)
    const bool live = (waveR < nTiles);
    const int pix0 = wave * 16;
    const int bb   = pix0 / (OH * OW);
    const int rem  = pix0 - bb * (OH * OW);
    const int oy   = rem / OW;
    const int ox0  = rem - oy * OW;
    const int half = lane >> 4;
    const int l15  = lane & 15;
    const int ox   = ox0 + l15;  // A row (pixel M=l15) for this lane

    v8f acc[NOC];
#pragma unroll
    for (int t = 0; t < NOC; ++t)
#pragma unroll
        for (int e = 0; e < 8; ++e) acc[t][e] = 0.0f;

    for (int pc0 = 0; pc0 < PAIRS; pc0 += CHUNK) {
        const int pn = (PAIRS - pc0 < CHUNK) ? (PAIRS - pc0) : CHUNK;
        __syncthreads();  // protect previous chunk's readers
        for (int p = threadIdx.x; p < pn * COUT; p += 256) {
            const int kp = p / COUT + pc0;
            const int oc = p - (p / COUT) * COUT;
            const int k0 = 2 * kp;
            v2f v;
            v.x = (oc < CVALID && k0 < KTOT) ? w[(size_t)oc * KTOT + k0] : 0.0f;
            v.y = (oc < CVALID && k0 + 1 < KTOT) ? w[(size_t)oc * KTOT + k0 + 1] : 0.0f;
            spair[p] = v;
        }
        __syncthreads();

        const int kkEnd = (pc0 + pn) * 2;
        for (int kk = pc0 * 2; kk < kkEnd; kk += 4) {
            const int ka = kk + 2 * half;  // A VGPR0 k; VGPR1 holds ka+1
            v2f a;
            a.x = (ka < KTOT) ? fetchA<CIN, STRIDE, INH, INW, UPS>(in, bb, ka, oy, ox) : 0.0f;
            a.y = (ka + 1 < KTOT) ? fetchA<CIN, STRIDE, INH, INW, UPS>(in, bb, ka + 1, oy, ox) : 0.0f;
            const int prow = (ka >> 1) - pc0;  // local pair index in LDS
#pragma unroll
            for (int t = 0; t < NOC; ++t) {
                const v2f bv = spair[prow * COUT + t * 16 + l15];
                acc[t] = wmma4(a, bv, acc[t]);
            }
        }
    }

    if (!live) return;
    // D layout: element r, lanes0-15 -> M=r, lanes16-31 -> M=r+8
#pragma unroll
    for (int t = 0; t < NOC; ++t) {
        const int oc = t * 16 + l15;
        if (oc >= CVALID) continue;
        const float bvs = bias[oc];
        float* po = out + ((size_t)bb * CVALID + oc) * ((size_t)OH * OW) +
                    (size_t)oy * OW + ox0 + 8 * half;
#pragma unroll
        for (int r = 0; r < 8; ++r) {
            float v = acc[t][r] + bvs;
            if (RELU) v = fmaxf(v, 0.0f);
            po[r] = v;
        }
    }
}

// ---------------------------------------------------------------------------
// 1x1 conv GEMM: NCHW activations -> row-major [Npix, COUT] (for enc conv3)
// Weights staged as k-pairs in static LDS (CIN*COUT*4 <= 16 KB here).
// ---------------------------------------------------------------------------
template <int CIN, int COUT, int NPIX, bool RELU>
__global__ __launch_bounds__(256) void conv1x1_nchw_to_rows(
    const float* __restrict__ in, const float* __restrict__ w,
    const float* __restrict__ bias, float* __restrict__ out, int nTiles) {
    constexpr int NOC = COUT / 16;
    __shared__ v2f spair[(CIN / 2) * COUT];
    const int lane  = threadIdx.x & 31;
    const int waveR = blockIdx.x * (blockDim.x >> 5) + (threadIdx.x >> 5);
    const int wave  = min(waveR, nTiles - 1);
    const bool live = (waveR < nTiles);
    const int pix0 = wave * 16;
    const int bb   = pix0 / NPIX;
    const int p0   = pix0 - bb * NPIX;
    const int half = lane >> 4;
    const int l15  = lane & 15;
    const float* pin = in + (size_t)bb * CIN * NPIX + p0 + l15;

    for (int p = threadIdx.x; p < (CIN / 2) * COUT; p += 256) {
        const int kp = p / COUT;
        const int oc = p - kp * COUT;
        v2f v;
        v.x = w[(size_t)oc * CIN + 2 * kp];
        v.y = w[(size_t)oc * CIN + 2 * kp + 1];
        spair[p] = v;
    }
    __syncthreads();

    v8f acc[NOC];
#pragma unroll
    for (int t = 0; t < NOC; ++t)
#pragma unroll
        for (int e = 0; e < 8; ++e) acc[t][e] = 0.0f;

    for (int kk = 0; kk < CIN; kk += 4) {
        const int ka = kk + 2 * half;
        v2f a;
        a.x = pin[(size_t)ka * NPIX];
        a.y = pin[(size_t)(ka + 1) * NPIX];
        const int prow = ka >> 1;
#pragma unroll
        for (int t = 0; t < NOC; ++t) {
            const v2f bv = spair[prow * COUT + t * 16 + l15];
            acc[t] = wmma4(a, bv, acc[t]);
        }
    }
    if (!live) return;
#pragma unroll
    for (int t = 0; t < NOC; ++t) {
        const int oc = t * 16 + l15;
        const float bvs = bias[oc];
#pragma unroll
        for (int r = 0; r < 8; ++r) {
            float v = acc[t][r] + bvs;
            if (RELU) v = fmaxf(v, 0.0f);
            out[(size_t)(pix0 + r + 8 * half) * COUT + oc] = v;
        }
    }
}

// ---------------------------------------------------------------------------
// 1x1 conv GEMM: row-major [Npix, CIN] -> NCHW (for dec conv1)
// ---------------------------------------------------------------------------
template <int CIN, int COUT, int NPIX, bool RELU>
__global__ __launch_bounds__(256) void conv1x1_rows_to_nchw(
    const float* __restrict__ in, const float* __restrict__ w,
    const float* __restrict__ bias, float* __restrict__ out, int nTiles) {
    constexpr int NOC = COUT / 16;
    __shared__ v2f spair[(CIN / 2) * COUT];
    const int lane  = threadIdx.x & 31;
    const int waveR = blockIdx.x * (blockDim.x >> 5) + (threadIdx.x >> 5);
    const int wave  = min(waveR, nTiles - 1);
    const bool live = (waveR < nTiles);
    const int pix0 = wave * 16;
    const int bb   = pix0 / NPIX;
    const int p0   = pix0 - bb * NPIX;
    const int half = lane >> 4;
    const int l15  = lane & 15;
    const float* pin = in + (size_t)(pix0 + l15) * CIN + 2 * half;

    for (int p = threadIdx.x; p < (CIN / 2) * COUT; p += 256) {
        const int kp = p / COUT;
        const int oc = p - kp * COUT;
        v2f v;
        v.x = w[(size_t)oc * CIN + 2 * kp];
        v.y = w[(size_t)oc * CIN + 2 * kp + 1];
        spair[p] = v;
    }
    __syncthreads();

    v8f acc[NOC];
#pragma unroll
    for (int t = 0; t < NOC; ++t)
#pragma unroll
        for (int e = 0; e < 8; ++e) acc[t][e] = 0.0f;

    for (int kk = 0; kk < CIN; kk += 4) {
        const int ka = kk + 2 * half;
        v2f a;
        a.x = pin[kk];      // = in[row*CIN + ka]
        a.y = pin[kk + 1];  // = in[row*CIN + ka + 1]
        const int prow = ka >> 1;
#pragma unroll
        for (int t = 0; t < NOC; ++t) {
            const v2f bv = spair[prow * COUT + t * 16 + l15];
            acc[t] = wmma4(a, bv, acc[t]);
        }
    }
    if (!live) return;
#pragma unroll
    for (int t = 0; t < NOC; ++t) {
        const int oc = t * 16 + l15;
        const float bvs = bias[oc];
        float* po = out + ((size_t)bb * COUT + oc) * NPIX + p0 + 8 * half;
#pragma unroll
        for (int r = 0; r < 8; ++r) {
            float v = acc[t][r] + bvs;
            if (RELU) v = fmaxf(v, 0.0f);
            po[r] = v;
        }
    }
}

// ---------------------------------------------------------------------------
// Codebook squared norms + zero the two loss slots
// ---------------------------------------------------------------------------
__global__ void k_cbnorm(const float* __restrict__ cb, float* __restrict__ cc,
                         float* __restrict__ losses) {
    const int i = blockIdx.x * blockDim.x + threadIdx.x;
    if (i < 512) {
        float s = 0.0f;
        for (int d = 0; d < 64; ++d) {
            const float v = cb[i * 64 + d];
            s += v * v;
        }
        cc[i] = s;
    }
    if (i == 0) { losses[0] = 0.0f; losses[1] = 0.0f; }
}

// ---------------------------------------------------------------------------
// VQ argmin: per wave, 16 rows of z vs all 512 codes.
// argmin_k (||c_k||^2 - 2 z.c_k)   (||z||^2 constant per row -> dropped)
// S-tile via WMMA f32 16x16x4, A (z) preloaded in registers, 32 col tiles.
// ---------------------------------------------------------------------------
__global__ __launch_bounds__(256) void vq_argmin(
    const float* __restrict__ z, const float* __restrict__ cb,
    const float* __restrict__ cc, int* __restrict__ idx, int nTiles) {
    const int lane = threadIdx.x & 31;
    const int wave = blockIdx.x * (blockDim.x >> 5) + (threadIdx.x >> 5);
    if (wave >= nTiles) return;
    const int row0 = wave * 16;
    const int half = lane >> 4;
    const int l15  = lane & 15;

    v2f A[16];  // K=64 -> 16 k-steps of 4
    const float* zp = z + (size_t)(row0 + l15) * 64 + 2 * half;
#pragma unroll
    for (int s = 0; s < 16; ++s) {
        A[s].x = zp[4 * s];
        A[s].y = zp[4 * s + 1];
    }

    float minv[8];
    int   mini[8];
#pragma unroll
    for (int r = 0; r < 8; ++r) { minv[r] = 3.4e38f; mini[r] = 0; }

    for (int ct = 0; ct < 32; ++ct) {
        const int col0 = ct * 16;
        const float* cp = cb + (size_t)(col0 + l15) * 64 + 2 * half;
        v8f acc;
#pragma unroll
        for (int e = 0; e < 8; ++e) acc[e] = 0.0f;
#pragma unroll
        for (int s = 0; s < 16; ++s) {
            v2f bv;
            bv.x = cp[4 * s];
            bv.y = cp[4 * s + 1];
            acc = wmma4(A[s], bv, acc);
        }
        const float ccv = cc[col0 + l15];
        const int   n   = col0 + l15;
#pragma unroll
        for (int r = 0; r < 8; ++r) {
            const float d = ccv - 2.0f * acc[r];
            if (d < minv[r]) { minv[r] = d; mini[r] = n; }
        }
    }

    // reduce argmin across the 16 lanes holding the same row
#pragma unroll
    for (int r = 0; r < 8; ++r) {
        float v = minv[r];
        int   i = mini[r];
        for (int off = 1; off < 16; off <<= 1) {
            const float ov = __shfl_xor(v, off, 32);
            const int   oi = __shfl_xor(i, off, 32);
            if (ov < v || (ov == v && oi < i)) { v = ov; i = oi; }
        }
        if (l15 == 0) idx[row0 + r + 8 * half] = i;
    }
}

// ---------------------------------------------------------------------------
// Gather zq = codebook[idx] and accumulate mean((zq - z)^2) into both losses
// ---------------------------------------------------------------------------
__global__ __launch_bounds__(256) void k_gather_loss(
    const float* __restrict__ z, const float* __restrict__ cb,
    const int* __restrict__ idx, float* __restrict__ zq,
    float* __restrict__ losses, float inv) {
    __shared__ float sred[8];
    const int i = blockIdx.x * 256 + threadIdx.x;
    const int n = i >> 6;
    const int d = i & 63;
    const float q = cb[(size_t)idx[n] * 64 + d];
    zq[i] = q;
    const float diff = q - z[i];
    float s = diff * diff;
#pragma unroll
    for (int off = 16; off > 0; off >>= 1) s += __shfl_xor(s, off, 32);
    const int lane = threadIdx.x & 31;
    const int w = threadIdx.x >> 5;
    if (lane == 0) sred[w] = s;
    __syncthreads();
    if (threadIdx.x == 0) {
        float t = 0.0f;
        for (int j = 0; j < 8; ++j) t += sred[j];
        atomicAdd(&losses[0], t * inv);
        atomicAdd(&losses[1], t * inv);
    }
}

// ---------------------------------------------------------------------------
extern "C" void kernel_launch(void* const* d_in, const int* in_sizes, int n_in,
                              void* d_out, int out_size, void* d_ws, size_t ws_size,
                              hipStream_t stream) {
    const float* x   = (const float*)d_in[0];
    const float* cb  = (const float*)d_in[1];
    const float* ew1 = (const float*)d_in[2];
    const float* eb1 = (const float*)d_in[3];
    const float* ew2 = (const float*)d_in[4];
    const float* eb2 = (const float*)d_in[5];
    const float* ew3 = (const float*)d_in[6];
    const float* eb3 = (const float*)d_in[7];
    const float* dw1 = (const float*)d_in[8];
    const float* db1 = (const float*)d_in[9];
    const float* dw2 = (const float*)d_in[10];
    const float* db2 = (const float*)d_in[11];
    const float* dw3 = (const float*)d_in[12];
    const float* db3 = (const float*)d_in[13];
    float* out = (float*)d_out;

    // workspace layout (floats); d2 aliases h1 (h1 dead after enc conv2)
    float* ws = (float*)d_ws;
    size_t off = 0;
    float* h1 = ws + off; off += (size_t)32 * 64 * 128 * 128;  // 33.55M
    float* h2 = ws + off; off += (size_t)32 * 32 * 64 * 64;    //  4.19M
    float* z  = ws + off; off += (size_t)131072 * 64;          //  8.39M
    float* zq = ws + off; off += (size_t)131072 * 64;          //  8.39M
    float* d1 = ws + off; off += (size_t)32 * 32 * 64 * 64;    //  4.19M
    float* cc = ws + off; off += 512;
    int*  idx = (int*)(ws + off); off += 131072;
    float* d2 = h1;

    // codebook norms + zero losses
    k_cbnorm<<<2, 256, 0, stream>>>(cb, cc, out);

    // encoder
    conv3x3_wmma<3, 64, 2, 256, 256, 128, 128, false, true, 64>
        <<<4096, 256, 0, stream>>>(x, ew1, eb1, h1, 32768);
    conv3x3_wmma<64, 32, 2, 128, 128, 64, 64, false, true, 32>
        <<<1024, 256, 0, stream>>>(h1, ew2, eb2, h2, 8192);
    conv1x1_nchw_to_rows<32, 64, 4096, false>
        <<<1024, 256, 0, stream>>>(h2, ew3, eb3, z, 8192);

    // vector quantization + losses
    vq_argmin<<<1024, 256, 0, stream>>>(z, cb, cc, idx, 8192);
    k_gather_loss<<<32768, 256, 0, stream>>>(z, cb, idx, zq, out,
                                             1.0f / 8388608.0f);

    // decoder (upsamples fused into conv A-fetch)
    conv1x1_rows_to_nchw<64, 32, 4096, true>
        <<<1024, 256, 0, stream>>>(zq, dw1, db1, d1, 8192);
    conv3x3_wmma<32, 64, 1, 64, 64, 128, 128, true, true, 64>
        <<<4096, 256, 0, stream>>>(d1, dw2, db2, d2, 32768);
    conv3x3_wmma<64, 16, 1, 128, 128, 256, 256, true, false, 3>
        <<<16384, 256, 0, stream>>>(d2, dw3, db3, out + 2, 131072);
}